// NoiseGenerator_66434554134561
// MI455X (gfx1250) — compile-verified
//
#include <hip/hip_runtime.h>

// GRU: B=256, T=2048, I=10, H=64. PyTorch gate order (r, z, n).
// Grid: 16 blocks (one 16-row batch tile). Block: 128 threads = 4 wave32's.
// Wave w owns hidden-column tile w (16 cols) and computes all three gates for
// it entirely in registers using V_WMMA_F32_16X16X4_F32 (exact f32 compute).
// h state ping-pongs through LDS (swizzled for b128 A-fragment loads) with a
// single SPLIT barrier per timestep (s_barrier_signal / s_barrier_wait) so
// global stores + next noise loads overlap the barrier wait.

typedef __attribute__((ext_vector_type(2))) float v2f;
typedef __attribute__((ext_vector_type(4))) float v4f;
typedef __attribute__((ext_vector_type(8))) float v8f;

#define GB 256
#define GT 2048
#define GI 10
#define GH 64

#define WMMA4(A, B, C) __builtin_amdgcn_wmma_f32_16x16x4_f32( \
    false, (A), false, (B), (short)0, (C), false, false)

// ---- fast transcendentals (hardware TRANS ops; guarded fallbacks) ----------
__device__ __forceinline__ float fast_rcp(float x) {
#if __has_builtin(__builtin_amdgcn_rcpf)
    return __builtin_amdgcn_rcpf(x);
#else
    return 1.0f / x;
#endif
}
__device__ __forceinline__ float fast_exp(float x) {
#if __has_builtin(__builtin_amdgcn_exp2f)
    return __builtin_amdgcn_exp2f(x * 1.4426950408889634f);
#else
    return __expf(x);
#endif
}
__device__ __forceinline__ float tanh_f(float x) {
#if __has_builtin(__builtin_amdgcn_tanhf)
    return __builtin_amdgcn_tanhf(x);          // gfx1250 v_tanh_f32
#else
    return 1.0f - 2.0f * fast_rcp(fast_exp(2.0f * x) + 1.0f);
#endif
}
__device__ __forceinline__ float sigmoid_f(float x) {
#if __has_builtin(__builtin_amdgcn_tanhf)
    return 0.5f + 0.5f * __builtin_amdgcn_tanhf(0.5f * x);
#else
    return fast_rcp(1.0f + fast_exp(-x));
#endif
}

// ---- split workgroup barrier (CDNA5 s_barrier_signal / s_barrier_wait) -----
#define BAR_SIGNAL() asm volatile("s_wait_dscnt 0x0\n\ts_barrier_signal -1" ::: "memory")
#define BAR_WAIT()   asm volatile("s_barrier_wait -1" ::: "memory")

// Noise A-fragments (K padded 10->12, three K=4 chunks).
// f32 A layout (16x4): lane L feeds row m=L%16, elems k = 4*kc + 2*(L/16)+{0,1}.
struct NFrag { v2f c[3]; };
__device__ __forceinline__ NFrag load_noise(const float* __restrict__ np, int lhi) {
    NFrag f;
    if (lhi == 0) {                 // features {0,1},{4,5},{8,9}
        f.c[0] = *(const v2f*)(np + 0);
        f.c[1] = *(const v2f*)(np + 4);
        f.c[2] = *(const v2f*)(np + 8);
    } else {                        // features {2,3},{6,7},{10,11->0}
        f.c[0] = *(const v2f*)(np + 2);
        f.c[1] = *(const v2f*)(np + 6);
        f.c[2] = (v2f){0.0f, 0.0f};
    }
    return f;
}

__global__ __launch_bounds__(128) void gru_scan_kernel(
    const float* __restrict__ noise,  // (B, T, I)
    const float* __restrict__ w_ih,   // (3H, I)
    const float* __restrict__ w_hh,   // (3H, H)
    const float* __restrict__ b_ih,   // (3H)
    const float* __restrict__ b_hh,   // (3H)
    float* __restrict__ out)          // (B, T, H)
{
    // Swizzled h state: [buf][m][lane-half][32]; lane (m,lhi) reads 32
    // contiguous floats = its 16 K-chunk fragments -> 8x ds_load_b128.
    __shared__ __align__(16) float sh_h[2][16][2][32];

    const int tid  = threadIdx.x;
    const int ct   = tid >> 5;          // column tile this wave owns (0..3)
    const int lane = tid & 31;
    const int lhi  = lane >> 4;
    const int lcol = lane & 15;
    const int b0   = blockIdx.x * 16;
    const int colg = ct * 16 + lcol;    // hidden column (0..63) this lane owns

    // ---- one-time: f32 weight B-fragments + biases into registers ----------
    // B layout (4x16): lane L supplies col n=L%16, elems k = 4*kc + 2*(L/16)+{0,1}.
    v2f bwih[3][3];
    v2f bwhh[3][16];
    float bgi[3];
    float bhn = 0.0f;

    for (int g = 0; g < 3; ++g) {
        const int col3 = g * GH + colg;
        for (int kc = 0; kc < 3; ++kc) {
            const int k0 = 4 * kc + 2 * lhi;
            v2f b;
            b[0] = (k0     < GI) ? w_ih[col3 * GI + k0]     : 0.0f;
            b[1] = (k0 + 1 < GI) ? w_ih[col3 * GI + k0 + 1] : 0.0f;
            bwih[g][kc] = b;
        }
        for (int kc = 0; kc < 16; ++kc) {
            const int k0 = 4 * kc + 2 * lhi;
            v2f b;
            b[0] = w_hh[col3 * GH + k0];
            b[1] = w_hh[col3 * GH + k0 + 1];
            bwhh[g][kc] = b;
        }
        bgi[g] = b_ih[col3] + ((g < 2) ? b_hh[col3] : 0.0f);
        if (g == 2) bhn = b_hh[col3];
    }
    v8f cgi[3], chn, czero;
    #pragma unroll
    for (int g = 0; g < 3; ++g)
        #pragma unroll
        for (int vv = 0; vv < 8; ++vv) cgi[g][vv] = bgi[g];
    #pragma unroll
    for (int vv = 0; vv < 8; ++vv) { chn[vv] = bhn; czero[vv] = 0.0f; }

    // ---- init h = 0 ---------------------------------------------------------
    float hreg[8];
    #pragma unroll
    for (int vv = 0; vv < 8; ++vv) hreg[vv] = 0.0f;
    for (int idx = tid; idx < 16 * 2 * 32; idx += 128)
        ((float*)sh_h[0])[idx] = 0.0f;
    BAR_SIGNAL();

    const int    m_a       = lane & 15;
    const size_t noise_row = (size_t)(b0 + m_a) * GT * GI;
    // where this lane's h value lands in the swizzled layout
    const int whalf = (colg >> 1) & 1;
    const int widx  = ((colg >> 2) << 1) | (colg & 1);

    NFrag an_next = load_noise(noise + noise_row, lhi);   // t = 0 (pipelined)

    for (int t = 0; t < GT; ++t) {
        BAR_WAIT();                       // h(t) published by all waves

        NFrag an = an_next;
        {   // software-pipeline next timestep's noise loads
            const size_t tn = (size_t)((t + 1 < GT) ? (t + 1) : t) * GI;
            an_next = load_noise(noise + noise_row + tn, lhi);
        }

        // ---- hidden-state A fragments: 32 contiguous floats per lane -------
        const float* __restrict__ hrow = &sh_h[t & 1][m_a][lhi][0];
        v2f ah[16];
        #pragma unroll
        for (int kc = 0; kc < 16; ++kc)
            ah[kc] = *((const v2f*)hrow + kc);

        // ---- WMMA chains: per gate, gi(3) + hidden split into 2 chains ------
        v8f accA[3], accB[3], gn;
        #pragma unroll
        for (int g = 0; g < 3; ++g) {
            v8f a = WMMA4(an.c[0], bwih[g][0], cgi[g]);
            a = WMMA4(an.c[1], bwih[g][1], a);
            a = WMMA4(an.c[2], bwih[g][2], a);
            if (g == 2) { gn = a; a = chn; }  // n: keep gi apart; b_hn seeds hidden
            #pragma unroll
            for (int kc = 0; kc < 8; ++kc)
                a = WMMA4(ah[kc], bwhh[g][kc], a);
            accA[g] = a;
            v8f bacc = czero;
            #pragma unroll
            for (int kc = 8; kc < 16; ++kc)
                bacc = WMMA4(ah[kc], bwhh[g][kc], bacc);
            accB[g] = bacc;
        }

        // ---- gate combine in registers; stage h stores ----------------------
        float hnew[8];
        float* __restrict__ hw = &sh_h[(t & 1) ^ 1][0][0][0];
        #pragma unroll
        for (int vv = 0; vv < 8; ++vv) {
            const int m = vv + 8 * lhi;           // C/D layout: M = v + 8*(L/16)
            const float r  = sigmoid_f(accA[0][vv] + accB[0][vv]);
            const float z  = sigmoid_f(accA[1][vv] + accB[1][vv]);
            const float hn = accA[2][vv] + accB[2][vv];      // h@w_hn^T + b_hn
            const float n  = tanh_f(gn[vv] + r * hn);
            const float h  = n + z * (hreg[vv] - n);         // (1-z)n + zh
            hreg[vv] = h;
            hnew[vv] = h;
            hw[(m * 2 + whalf) * 32 + widx] = h;             // swizzled store
        }
        BAR_SIGNAL();   // dscnt-drained signal; overlap stores/loads with wait

        #pragma unroll
        for (int vv = 0; vv < 8; ++vv) {
            const int m = vv + 8 * lhi;
            out[((size_t)(b0 + m) * GT + t) * GH + colg] = hnew[vv];
        }
    }
}

extern "C" void kernel_launch(void* const* d_in, const int* in_sizes, int n_in,
                              void* d_out, int out_size, void* d_ws, size_t ws_size,
                              hipStream_t stream) {
    (void)in_sizes; (void)n_in; (void)out_size; (void)d_ws; (void)ws_size;
    const float* noise = (const float*)d_in[0];
    const float* w_ih  = (const float*)d_in[1];
    const float* w_hh  = (const float*)d_in[2];
    const float* b_ih  = (const float*)d_in[3];
    const float* b_hh  = (const float*)d_in[4];
    float* out = (float*)d_out;

    dim3 grid(GB / 16);   // 16 batch tiles
    dim3 block(128);      // 4 wave32's: one hidden-column tile each
    gru_scan_kernel<<<grid, block, 0, stream>>>(noise, w_ih, w_hh, b_ih, b_hh, out);
}